// LatticeGNN_87247965651265
// MI455X (gfx1250) — compile-verified
//
#include <hip/hip_runtime.h>

// Problem constants (match reference setup_inputs()).
#define NN   50000     // nodes
#define EE   800000    // edges
#define HDIM 64        // channels (C_IN == H == 64)
#define SLOPE 0.01f

typedef float v2f __attribute__((ext_vector_type(2)));
typedef float v8f __attribute__((ext_vector_type(8)));

// ---------------------------------------------------------------- utilities
__global__ void zero_f32_kernel(float* __restrict__ p, int n) {
    int i = blockIdx.x * blockDim.x + threadIdx.x;
    if (i < n) p[i] = 0.0f;
}

__global__ void zero_i32_kernel(int* __restrict__ p, int n) {
    int i = blockIdx.x * blockDim.x + threadIdx.x;
    if (i < n) p[i] = 0;
}

// ------------------------------------------------------------------ degrees
__global__ void degree_kernel(const int* __restrict__ dst, int* __restrict__ deg) {
    int e = blockIdx.x * blockDim.x + threadIdx.x;
    if (e < EE) atomicAdd(&deg[dst[e]], 1);
}

__global__ void invdeg_kernel(const int* __restrict__ deg, float* __restrict__ invd) {
    int i = blockIdx.x * blockDim.x + threadIdx.x;
    if (i < NN) {
        int d = deg[i];
        invd[i] = 1.0f / (float)(d > 1 ? d : 1);
    }
}

// --------------------------------------------- weight pre-swizzle (one-shot)
// Rewrites a 64x64 row-major weight matrix into per-lane WMMA B-fragment
// order: frag[kk][n][lane] = { W[k][col], W[k+1][col] } with
//   k = 4*kk + 2*(lane>=16), col = 16*n + (lane&15).
// Lets the GEMM inner loop fetch each fragment as ONE ds_load_b64 with an
// immediate offset (no repacking movs, no address arithmetic).
__global__ void swizzle_w_kernel(const float* __restrict__ W,
                                 float2* __restrict__ sw) {
    int idx = blockIdx.x * blockDim.x + threadIdx.x;  // 0..2047
    if (idx >= 2048) return;
    int lane = idx & 31;
    int n    = (idx >> 5) & 3;
    int kk   = idx >> 7;
    int k    = kk * 4 + ((lane >> 4) << 1);
    int col  = n * 16 + (lane & 15);
    sw[idx] = make_float2(W[k * HDIM + col], W[(k + 1) * HDIM + col]);
}

// ------------------------------------------------------- neighbor scatter-add
// One thread per (edge, 16-channel chunk): float4 gathers from h[src],
// native fp32 global atomics onto agg[dst]. Working set lives in L2 (192MB).
__global__ void scatter_kernel(const float* __restrict__ h,
                               const int* __restrict__ src,
                               const int* __restrict__ dst,
                               float* __restrict__ agg) {
    int idx = blockIdx.x * blockDim.x + threadIdx.x;
    if (idx >= EE * 4) return;
    int e  = idx >> 2;
    int c0 = (idx & 3) << 4;           // 0,16,32,48
    int s = src[e];
    int d = dst[e];
    const float4* hp = (const float4*)(h + s * HDIM + c0);
    float* ap = agg + d * HDIM + c0;
#pragma unroll
    for (int q = 0; q < 4; ++q) {
        float4 v = hp[q];
        unsafeAtomicAdd(ap + q * 4 + 0, v.x);   // global_atomic_add_f32
        unsafeAtomicAdd(ap + q * 4 + 1, v.y);
        unsafeAtomicAdd(ap + q * 4 + 2, v.z);
        unsafeAtomicAdd(ap + q * 4 + 3, v.w);
    }
}

// ------------------------------------------------- fused SAGE GEMM via WMMA
// out = leaky_relu( (inv_deg * agg) @ Wl + bl + h @ Wr )
// 256 threads = 8 wave32s; each wave computes a 16x64 output tile with
// v_wmma_f32_16x16x4_f32. Weights arrive pre-swizzled in fragment order.
__global__ __launch_bounds__(256)
void sage_gemm_kernel(const float* __restrict__ agg,
                      const float* __restrict__ hin,
                      const float* __restrict__ swzWl,   // 2048 float2, fragment order
                      const float* __restrict__ bl,
                      const float* __restrict__ swzWr,   // 2048 float2, fragment order
                      const float* __restrict__ invd,
                      float* __restrict__ hout) {
    __shared__ v2f   sW[2 * 2048];    // [0..2047]=Wl frags, [2048..4095]=Wr frags
    __shared__ float sbl[HDIM];

    int t = threadIdx.x;
    // Coalesced float4 staging of both pre-swizzled matrices (32 KB total).
    {
        const float4* gl = (const float4*)swzWl;   // 1024 float4
        const float4* gr = (const float4*)swzWr;
        float4* s4 = (float4*)sW;
#pragma unroll
        for (int i = 0; i < 4; ++i) {
            s4[t + 256 * i]        = gl[t + 256 * i];
            s4[1024 + t + 256 * i] = gr[t + 256 * i];
        }
        if (t < HDIM) sbl[t] = bl[t];
    }
    __syncthreads();

    int wave = t >> 5;
    int lane = t & 31;
    int mcol  = lane & 15;          // A: row index / B,C: column index
    int khalf = (lane >> 4) << 1;   // 0 for lanes 0-15, 2 for lanes 16-31

    int r0 = blockIdx.x * 128 + wave * 16;   // 16-row tile base
    int r  = r0 + mcol;
    int rc = r < NN ? r : (NN - 1);          // clamp (EXEC must stay all-1)
    float scale = invd[rc];

    const float* arow = agg + rc * HDIM;
    const float* hrow = hin + rc * HDIM;

    // Per-lane fragment base pointers; all inner-loop offsets are immediates.
    const v2f* fl = sW + lane;           // Wl fragments, stride 32 per (kk,n)
    const v2f* fr = sW + 2048 + lane;    // Wr fragments

    // Accumulators: 4 column tiles of 16; bias broadcast per column.
    v8f acc[4];
#pragma unroll
    for (int n = 0; n < 4; ++n) {
        float b = sbl[n * 16 + mcol];
#pragma unroll
        for (int v = 0; v < 8; ++v) acc[n][v] = b;
    }

#pragma unroll
    for (int kk = 0; kk < 16; ++kk) {
        int k = kk * 4 + khalf;
        v2f aM, aH;
        aM[0] = arow[k]     * scale;
        aM[1] = arow[k + 1] * scale;
        aH[0] = hrow[k];
        aH[1] = hrow[k + 1];
#pragma unroll
        for (int n = 0; n < 4; ++n) {
            v2f bL = fl[kk * 128 + n * 32];   // single ds_load_b64, imm offset
            v2f bR = fr[kk * 128 + n * 32];
            acc[n] = __builtin_amdgcn_wmma_f32_16x16x4_f32(
                false, aM, false, bL, (short)0, acc[n], false, false);
            acc[n] = __builtin_amdgcn_wmma_f32_16x16x4_f32(
                false, aH, false, bR, (short)0, acc[n], false, false);
        }
    }

    // Fused leaky-ReLU + store. C/D layout: N = lane&15, M = v + 8*(lane>=16).
    int mbase = (lane >> 4) * 8;
#pragma unroll
    for (int n = 0; n < 4; ++n) {
        int col = n * 16 + mcol;
#pragma unroll
        for (int v = 0; v < 8; ++v) {
            int row = r0 + mbase + v;
            if (row < NN) {
                float x = acc[n][v];
                hout[row * HDIM + col] = x > 0.0f ? x : x * SLOPE;
            }
        }
    }
}

// ------------------------------------------------------------- output head
__global__ void out_proj_kernel(const float* __restrict__ h,
                                const float* __restrict__ Wout,
                                const float* __restrict__ bout,
                                float* __restrict__ out) {
    int i = blockIdx.x * blockDim.x + threadIdx.x;
    if (i >= NN) return;
    const float4* hp = (const float4*)(h + i * HDIM);
    const float4* wp = (const float4*)Wout;
    float s = 0.0f;
#pragma unroll
    for (int q = 0; q < 16; ++q) {
        float4 a = hp[q];
        float4 b = wp[q];
        s += a.x * b.x + a.y * b.y + a.z * b.z + a.w * b.w;
    }
    out[i] = s + bout[0];
}

// ------------------------------------------------------------------- launch
extern "C" void kernel_launch(void* const* d_in, const int* in_sizes, int n_in,
                              void* d_out, int out_size, void* d_ws, size_t ws_size,
                              hipStream_t stream) {
    const float* x   = (const float*)d_in[0];
    const int*   ei  = (const int*)d_in[1];
    const int*   src = ei;              // edge_index[0]
    const int*   dst = ei + EE;         // edge_index[1]
    const float* Wl[3] = { (const float*)d_in[2], (const float*)d_in[5], (const float*)d_in[8]  };
    const float* bl[3] = { (const float*)d_in[3], (const float*)d_in[6], (const float*)d_in[9]  };
    const float* Wr[3] = { (const float*)d_in[4], (const float*)d_in[7], (const float*)d_in[10] };
    const float* Wout  = (const float*)d_in[11];
    const float* bout  = (const float*)d_in[12];

    // Workspace (~38.9 MB):
    //   invd[N] | deg[N] | agg[N*64] | h1[N*64] | h2[N*64] | swz[6*4096]
    float* ws   = (float*)d_ws;
    float* invd = ws;
    int*   deg  = (int*)(ws + NN);
    float* agg  = ws + 2 * NN;
    float* h1   = agg + NN * HDIM;
    float* h2   = h1  + NN * HDIM;
    float* swz  = h2  + NN * HDIM;      // 6 matrices x 2048 float2
    float* swzL[3] = { swz,            swz + 1 * 4096, swz + 2 * 4096 };
    float* swzR[3] = { swz + 3 * 4096, swz + 4 * 4096, swz + 5 * 4096 };

    const int T = 256;

    // One-shot weight pre-swizzle into WMMA fragment order.
    for (int l = 0; l < 3; ++l) {
        swizzle_w_kernel<<<8, T, 0, stream>>>(Wl[l], (float2*)swzL[l]);
        swizzle_w_kernel<<<8, T, 0, stream>>>(Wr[l], (float2*)swzR[l]);
    }

    zero_i32_kernel<<<(NN + T - 1) / T, T, 0, stream>>>(deg, NN);
    degree_kernel<<<(EE + T - 1) / T, T, 0, stream>>>(dst, deg);
    invdeg_kernel<<<(NN + T - 1) / T, T, 0, stream>>>(deg, invd);

    const float* hin = x;
    float* houts[3] = { h1, h2, h1 };
    for (int l = 0; l < 3; ++l) {
        zero_f32_kernel<<<(NN * HDIM + T - 1) / T, T, 0, stream>>>(agg, NN * HDIM);
        scatter_kernel<<<(EE * 4 + T - 1) / T, T, 0, stream>>>(hin, src, dst, agg);
        sage_gemm_kernel<<<(NN + 127) / 128, T, 0, stream>>>(
            agg, hin, swzL[l], bl[l], swzR[l], invd, houts[l]);
        hin = houts[l];
    }

    out_proj_kernel<<<(NN + T - 1) / T, T, 0, stream>>>(h1, Wout, bout, (float*)d_out);
}